// KernelApply_40973988004096
// MI455X (gfx1250) — compile-verified
//
#include <hip/hip_runtime.h>
#include <math.h>

// Problem constants (fixed by the reference's setup_inputs()).
#define BB 4
#define CC 3
#define HH 720
#define WW 1280
#define KK 5
#define RR 2              // KK/2
// Tile: 64 wide x 4 high, one thread per pixel, 256 threads = 8 wave32.
#define TW 64
#define TH 4
#define LH (TH + 2 * RR)  // 8 rows incl. halo
// Staged row: 80 floats starting at global col (x0 - 8) so every b128 chunk
// (16 B) is aligned. Needed halo is only x0-2 .. x0+65; extra cols are slack.
#define LW2 80
#define HALO_LEFT 8       // staged-origin offset: global col = x0 - HALO_LEFT
#define NCHUNK (LW2 / 4)  // 20 b128 chunks per staged row
#define NTHREADS (TW * TH)

typedef int v4i __attribute__((ext_vector_type(4)));
typedef __attribute__((address_space(1))) int* gptr_i32;
typedef __attribute__((address_space(3))) int* lptr_i32;
typedef __attribute__((address_space(1))) v4i* gptr_v4i;
typedef __attribute__((address_space(3))) v4i* lptr_v4i;

// ---- CDNA5 async global->LDS copies (ASYNCcnt-tracked, bypass VGPRs) ----
__device__ __forceinline__ void async_copy_b32(const float* g, float* l) {
#if __has_builtin(__builtin_amdgcn_global_load_async_to_lds_b32)
  __builtin_amdgcn_global_load_async_to_lds_b32(
      (gptr_i32)(void*)g, (lptr_i32)(void*)l, 0, 0);
#else
  *l = *g;
#endif
}

__device__ __forceinline__ void async_copy_b128(const float* g, float* l) {
#if __has_builtin(__builtin_amdgcn_global_load_async_to_lds_b128)
  __builtin_amdgcn_global_load_async_to_lds_b128(
      (gptr_v4i)(void*)g, (lptr_v4i)(void*)l, 0, 0);
#else
  async_copy_b32(g + 0, l + 0);
  async_copy_b32(g + 1, l + 1);
  async_copy_b32(g + 2, l + 2);
  async_copy_b32(g + 3, l + 3);
#endif
}

__device__ __forceinline__ void wait_async0() {
#if __has_builtin(__builtin_amdgcn_s_wait_asynccnt)
  __builtin_amdgcn_s_wait_asynccnt(0);
#else
  asm volatile("s_wait_asynccnt 0" ::: "memory");
#endif
}

__global__ __launch_bounds__(NTHREADS) void kernel_weighting_f32(
    const float* __restrict__ data,   // [B, C, H, W]
    const float* __restrict__ kern,   // [B, K*K, H, W]
    float* __restrict__ out) {        // [B, C, H, W]
  __shared__ __align__(16) float s_data[CC][LH][LW2];

  const int tid = threadIdx.x;
  const int tx  = tid & (TW - 1);
  const int ty  = tid >> 6;          // tid / TW
  const int x0  = blockIdx.x * TW;
  const int y0  = blockIdx.y * TH;
  const int b   = blockIdx.z;
  const int x   = x0 + tx;
  const int y   = y0 + ty;

  // ------------- Phase 1: async b128-stage data tile (+halo) into LDS ------
  // Rows are clamped vertically; horizontal chunks are clamped into [0, W-4].
  // Chunks never straddle the validity boundary (cols 0 and W are 4-aligned),
  // and clamped "don't-care" chunks feed only taps whose weight is zeroed.
  const float* dbase = data + (size_t)b * CC * HH * WW;
#pragma unroll
  for (int it = 0; it < 2; ++it) {
    const int i = tid + it * NTHREADS;            // chunk id, 480 total
    if (i < CC * LH * NCHUNK) {
      const int c   = i / (LH * NCHUNK);          // /160 (const: mul+shift)
      const int rem = i - c * (LH * NCHUNK);
      const int row = rem / NCHUNK;               // /20
      const int k   = rem - row * NCHUNK;
      int gy = y0 - RR + row;
      gy = gy < 0 ? 0 : (gy > HH - 1 ? HH - 1 : gy);
      int gc = x0 - HALO_LEFT + 4 * k;
      gc = gc < 0 ? 0 : (gc > WW - 4 ? WW - 4 : gc);
      async_copy_b128(dbase + ((size_t)c * HH + gy) * WW + gc,
                      &s_data[c][row][4 * k]);
    }
  }

  // ------------- Phase 2 (overlaps DMA): load + softmax the 25 weights -----
  // kernels is a 369 MB read-once stream (> L2) -> non-temporal loads.
  const float* kbase = kern + ((size_t)b * (KK * KK) * HH + y) * WW + x;
  float w[KK * KK];
  float m = -INFINITY;
#pragma unroll
  for (int k = 0; k < KK * KK; ++k) {
    w[k] = __builtin_nontemporal_load(kbase + (size_t)k * HH * WW);
    m = fmaxf(m, w[k]);
  }
  float s = 0.0f;
#pragma unroll
  for (int k = 0; k < KK * KK; ++k) {
    w[k] = __expf(w[k] - m);
    s += w[k];
  }
  const float inv = 1.0f / s;
#pragma unroll
  for (int i = 0; i < KK; ++i) {
#pragma unroll
    for (int j = 0; j < KK; ++j) {
      const int yy = y + i - RR;
      const int xx = x + j - RR;
      const bool ok = (yy >= 0) && (yy < HH) && (xx >= 0) && (xx < WW);
      w[i * KK + j] = ok ? w[i * KK + j] * inv : 0.0f;  // zero-pad semantics
    }
  }

  // ------------- Phase 3: wait for DMA, then 5x5 weighted stencil ----------
  wait_async0();     // each wave waits for its own async transfers
  __syncthreads();   // all waves see the full tile

  // LDS col for tap j of pixel tx: (x + j - 2) - (x0 - 8) = tx + j + 6.
  float acc0 = 0.0f, acc1 = 0.0f, acc2 = 0.0f;
#pragma unroll
  for (int i = 0; i < KK; ++i) {
#pragma unroll
    for (int j = 0; j < KK; ++j) {
      const float wk = w[i * KK + j];
      acc0 = fmaf(wk, s_data[0][ty + i][tx + j + HALO_LEFT - RR], acc0);
      acc1 = fmaf(wk, s_data[1][ty + i][tx + j + HALO_LEFT - RR], acc1);
      acc2 = fmaf(wk, s_data[2][ty + i][tx + j + HALO_LEFT - RR], acc2);
    }
  }

  // Write-once output -> non-temporal stores.
  float* obase = out + (((size_t)b * CC) * HH + y) * WW + x;
  __builtin_nontemporal_store(acc0, obase);
  __builtin_nontemporal_store(acc1, obase + (size_t)HH * WW);
  __builtin_nontemporal_store(acc2, obase + 2 * (size_t)HH * WW);
}

extern "C" void kernel_launch(void* const* d_in, const int* in_sizes, int n_in,
                              void* d_out, int out_size, void* d_ws, size_t ws_size,
                              hipStream_t stream) {
  (void)in_sizes; (void)n_in; (void)out_size; (void)d_ws; (void)ws_size;
  const float* data = (const float*)d_in[0];
  const float* kern = (const float*)d_in[1];
  float* out        = (float*)d_out;

  dim3 grid(WW / TW, HH / TH, BB);   // 20 x 180 x 4
  dim3 block(NTHREADS);
  kernel_weighting_f32<<<grid, block, 0, stream>>>(data, kern, out);
}